// GAT_37709812858998
// MI455X (gfx1250) — compile-verified
//
#include <hip/hip_runtime.h>
#include <hip/hip_bf16.h>

typedef float v2f __attribute__((ext_vector_type(2)));
typedef float v8f __attribute__((ext_vector_type(8)));

#define NEG_SLOPE 0.2f
#define BN_EPS 1e-5f

// ---------------------------------------------------------------------------
// helpers
// ---------------------------------------------------------------------------
__device__ __forceinline__ float atomicMaxFloat(float* addr, float v) {
  // monotonic int/uint trick: works for mixed-sign floats
  if (v >= 0.0f)
    return __int_as_float(atomicMax((int*)addr, __float_as_int(v)));
  else
    return __uint_as_float(atomicMin((unsigned int*)addr, __float_as_uint(v)));
}

__device__ __forceinline__ void edge_sd(const int* __restrict__ ei, int E, int e,
                                        int& s, int& d) {
  if (e < E) { s = ei[e]; d = ei[E + e]; }
  else       { s = e - E; d = e - E; }      // implicit self-loops appended
}

// ---------------------------------------------------------------------------
// Fused dual GEMM:  Ha = X @ Wa, Hb = X @ Wb   (H fixed = 128, fp32 WMMA)
// One wave computes a 16-row strip x all 8 column tiles of ONE output matrix,
// reusing the A fragment across the 8 WMMAs of each K-step.
// ---------------------------------------------------------------------------
__global__ __launch_bounds__(256)
void gat_gemm_dual(const float* __restrict__ X,
                   const float* __restrict__ Wa,
                   const float* __restrict__ Wb,
                   float* __restrict__ Ha,
                   float* __restrict__ Hb,
                   int N, int K) {
  const int H = 128;
  int lane = threadIdx.x & 31;
  int wid  = blockIdx.x * (blockDim.x >> 5) + (threadIdx.x >> 5);
  int rowTiles = (N + 15) >> 4;
  if (wid >= rowTiles * 2) return;          // wave-uniform: EXEC stays full
  int rowBase = (wid >> 1) << 4;
  const float* W    = (wid & 1) ? Wb : Wa;
  float*       Hout = (wid & 1) ? Hb : Ha;

  int m = rowBase + (lane & 15);
  if (m >= N) m = N - 1;                    // clamp (N is a multiple of 16 here)
  int khalf = (lane >> 4) << 1;             // 0 for lanes 0-15, 2 for 16-31
  int n0    = lane & 15;

  v8f zero = {};
  v8f acc[8];
#pragma unroll
  for (int t = 0; t < 8; ++t) acc[t] = zero;

  for (int k = 0; k < K; k += 4) {
    v2f a;
    const float* xp = X + (size_t)m * K + (k + khalf);
    a.x = xp[0];
    a.y = xp[1];
    const float* wp = W + (size_t)(k + khalf) * H + n0;
#pragma unroll
    for (int t = 0; t < 8; ++t) {
      v2f b;
      b.x = wp[t * 16];
      b.y = wp[H + t * 16];
      acc[t] = __builtin_amdgcn_wmma_f32_16x16x4_f32(
          false, a, false, b, (short)0, acc[t], false, false);
    }
  }

  int rowOff = (lane >> 4) << 3;            // C/D layout: lanes 16-31 -> M+8
#pragma unroll
  for (int t = 0; t < 8; ++t) {
#pragma unroll
    for (int r = 0; r < 8; ++r) {
      int row = rowBase + rowOff + r;
      if (row < N) Hout[(size_t)row * H + t * 16 + n0] = acc[t][r];
    }
  }
}

// ---------------------------------------------------------------------------
// Generic WMMA GEMM with bias:  Out[N,Hout] = X[N,K] @ W[K,Hout] + bias
// HT = number of 16-wide column tiles (compile-time); cols masked to Hout.
// ---------------------------------------------------------------------------
template <int HT>
__global__ __launch_bounds__(256)
void gemm_bias_wmma(const float* __restrict__ X,
                    const float* __restrict__ W,
                    const float* __restrict__ bias,
                    float* __restrict__ Out,
                    int N, int K, int Hout) {
  int lane = threadIdx.x & 31;
  int wid  = blockIdx.x * (blockDim.x >> 5) + (threadIdx.x >> 5);
  int rowTiles = (N + 15) >> 4;
  if (wid >= rowTiles) return;
  int rowBase = wid << 4;

  int m = rowBase + (lane & 15);
  if (m >= N) m = N - 1;
  int khalf = (lane >> 4) << 1;
  int n0    = lane & 15;

  v8f zero = {};
  v8f acc[HT];
#pragma unroll
  for (int t = 0; t < HT; ++t) acc[t] = zero;

  for (int k = 0; k < K; k += 4) {
    v2f a;
    const float* xp = X + (size_t)m * K + (k + khalf);
    a.x = xp[0];
    a.y = xp[1];
    const float* wp = W + (size_t)(k + khalf) * Hout;
#pragma unroll
    for (int t = 0; t < HT; ++t) {
      int n = t * 16 + n0;
      v2f b;
      b.x = (n < Hout) ? wp[n] : 0.0f;
      b.y = (n < Hout) ? wp[Hout + n] : 0.0f;
      acc[t] = __builtin_amdgcn_wmma_f32_16x16x4_f32(
          false, a, false, b, (short)0, acc[t], false, false);
    }
  }

  int rowOff = (lane >> 4) << 3;
#pragma unroll
  for (int t = 0; t < HT; ++t) {
    int n = t * 16 + n0;
    float bs = (n < Hout) ? bias[n] : 0.0f;
#pragma unroll
    for (int r = 0; r < 8; ++r) {
      int row = rowBase + rowOff + r;
      if (row < N && n < Hout) Out[(size_t)row * Hout + n] = acc[t][r] + bs;
    }
  }
}

// ---------------------------------------------------------------------------
// Per-node attention dots: a_src[n] = h_src[n,:]@att_src, same for dst.
// One wave per node; H = 128.
// ---------------------------------------------------------------------------
__global__ __launch_bounds__(256)
void gat_dots(const float* __restrict__ Hs, const float* __restrict__ Hd,
              const float* __restrict__ atts, const float* __restrict__ attd,
              float* __restrict__ as_, float* __restrict__ ad_, int N, int H) {
  int lane = threadIdx.x & 31;
  int wid  = blockIdx.x * (blockDim.x >> 5) + (threadIdx.x >> 5);
  if (wid >= N) return;
  float ss = 0.0f, sd = 0.0f;
  for (int j = lane; j < H; j += 32) {
    ss += Hs[(size_t)wid * H + j] * atts[j];
    sd += Hd[(size_t)wid * H + j] * attd[j];
  }
#pragma unroll
  for (int off = 16; off > 0; off >>= 1) {
    ss += __shfl_xor(ss, off, 32);
    sd += __shfl_xor(sd, off, 32);
  }
  if (lane == 0) { as_[wid] = ss; ad_[wid] = sd; }
}

// ---------------------------------------------------------------------------
// small init / elementwise kernels
// ---------------------------------------------------------------------------
__global__ void init_stats_kernel(float* mx, float* den, int N) {
  int i = blockIdx.x * blockDim.x + threadIdx.x;
  if (i < N) { mx[i] = -3.0e38f; den[i] = 0.0f; }
}

__global__ void fill_bias_kernel(float* __restrict__ out,
                                 const float* __restrict__ bias,
                                 long long total, int H) {
  long long i = (long long)blockIdx.x * blockDim.x + threadIdx.x;
  long long stride = (long long)gridDim.x * blockDim.x;
  for (; i < total; i += stride) out[i] = bias[(int)(i % H)];
}

__global__ void zero_kernel(float* p, int n) {
  int i = blockIdx.x * blockDim.x + threadIdx.x;
  if (i < n) p[i] = 0.0f;
}

__global__ void relu_kernel(float* p, long long n) {
  long long i = (long long)blockIdx.x * blockDim.x + threadIdx.x;
  long long stride = (long long)gridDim.x * blockDim.x;
  for (; i < n; i += stride) p[i] = fmaxf(p[i], 0.0f);
}

// ---------------------------------------------------------------------------
// Edge phase (segment softmax + scatter)
// ---------------------------------------------------------------------------
__global__ void edge_scores_kernel(const int* __restrict__ ei,
                                   const float* __restrict__ as_,
                                   const float* __restrict__ ad_,
                                   float* __restrict__ esc,
                                   float* __restrict__ mx, int E, int Et) {
  int e = blockIdx.x * blockDim.x + threadIdx.x;
  if (e >= Et) return;
  int s, d;
  edge_sd(ei, E, e, s, d);
  float v = as_[s] + ad_[d];
  v = (v > 0.0f) ? v : NEG_SLOPE * v;       // leaky_relu
  esc[e] = v;
  atomicMaxFloat(&mx[d], v);
}

__global__ void edge_exp_kernel(const int* __restrict__ ei,
                                float* __restrict__ esc,
                                const float* __restrict__ mx,
                                float* __restrict__ den, int E, int Et) {
  int e = blockIdx.x * blockDim.x + threadIdx.x;
  if (e >= Et) return;
  int s, d;
  edge_sd(ei, E, e, s, d);
  float p = expf(esc[e] - mx[d]);
  esc[e] = p;
  atomicAdd(&den[d], p);
}

// one wave per edge; H = 128 -> each lane moves a float4
__global__ __launch_bounds__(256)
void edge_scatter_kernel(const int* __restrict__ ei,
                         const float* __restrict__ esc,
                         const float* __restrict__ den,
                         const float* __restrict__ Hs,
                         float* __restrict__ out, int E, int Et) {
  int lane = threadIdx.x & 31;
  int e    = blockIdx.x * (blockDim.x >> 5) + (threadIdx.x >> 5);
  if (e >= Et) return;
  int s, d;
  edge_sd(ei, E, e, s, d);
  float alpha = esc[e] / den[d];
  const float4* hp = (const float4*)(Hs + (size_t)s * 128);
  float4 v = hp[lane];
  float* op = out + (size_t)d * 128 + lane * 4;
  atomicAdd(op + 0, alpha * v.x);
  atomicAdd(op + 1, alpha * v.y);
  atomicAdd(op + 2, alpha * v.z);
  atomicAdd(op + 3, alpha * v.w);
}

// ---------------------------------------------------------------------------
// BatchNorm over nodes (per-column stats), then fused affine+relu
// ---------------------------------------------------------------------------
__global__ void bn_partial_kernel(const float* __restrict__ h,
                                  float* __restrict__ sum,
                                  float* __restrict__ sumsq, int N, int FC) {
  int c = threadIdx.x;                       // blockDim.x == FC
  float s = 0.0f, q = 0.0f;
  for (int r = blockIdx.x; r < N; r += gridDim.x) {
    float v = h[(size_t)r * FC + c];
    s += v;
    q += v * v;
  }
  atomicAdd(&sum[c], s);
  atomicAdd(&sumsq[c], q);
}

__global__ void bn_finalize_kernel(const float* __restrict__ sum,
                                   const float* __restrict__ sumsq,
                                   const float* __restrict__ gamma,
                                   const float* __restrict__ beta,
                                   float* __restrict__ scale,
                                   float* __restrict__ shift, int N, int FC) {
  int c = threadIdx.x;
  if (c >= FC) return;
  float inv = 1.0f / (float)N;
  float mu  = sum[c] * inv;
  float var = sumsq[c] * inv - mu * mu;
  float rstd = rsqrtf(var + BN_EPS);
  float sc = gamma[c] * rstd;
  scale[c] = sc;
  shift[c] = beta[c] - mu * sc;
}

__global__ void bn_apply_relu_kernel(float* __restrict__ h,
                                     const float* __restrict__ scale,
                                     const float* __restrict__ shift,
                                     long long total, int FC) {
  long long i = (long long)blockIdx.x * blockDim.x + threadIdx.x;
  long long stride = (long long)gridDim.x * blockDim.x;
  for (; i < total; i += stride) {
    int c = (int)(i % FC);
    h[i] = fmaxf(h[i] * scale[c] + shift[c], 0.0f);
  }
}

// ---------------------------------------------------------------------------
// Row-wise log_softmax (in place), C <= 64. One wave per row.
// ---------------------------------------------------------------------------
__global__ __launch_bounds__(256)
void log_softmax_kernel(float* __restrict__ out, int N, int C) {
  int lane = threadIdx.x & 31;
  int wid  = blockIdx.x * (blockDim.x >> 5) + (threadIdx.x >> 5);
  if (wid >= N) return;
  float* row = out + (size_t)wid * C;
  float v0 = (lane < C) ? row[lane] : -3.0e38f;
  float v1 = (lane + 32 < C) ? row[lane + 32] : -3.0e38f;
  float mx = fmaxf(v0, v1);
#pragma unroll
  for (int off = 16; off > 0; off >>= 1) mx = fmaxf(mx, __shfl_xor(mx, off, 32));
  float s = ((lane < C) ? expf(v0 - mx) : 0.0f) +
            ((lane + 32 < C) ? expf(v1 - mx) : 0.0f);
#pragma unroll
  for (int off = 16; off > 0; off >>= 1) s += __shfl_xor(s, off, 32);
  float lse = mx + logf(s);
  if (lane < C) row[lane] = v0 - lse;
  if (lane + 32 < C) row[lane + 32] = v1 - lse;
}

// ---------------------------------------------------------------------------
// host side
// ---------------------------------------------------------------------------
static inline int divup(long long a, long long b) { return (int)((a + b - 1) / b); }

extern "C" void kernel_launch(void* const* d_in, const int* in_sizes, int n_in,
                              void* d_out, int out_size, void* d_ws, size_t ws_size,
                              hipStream_t stream) {
  const float* x     = (const float*)d_in[0];
  const int*   ei    = (const int*)d_in[1];
  const float* W1s   = (const float*)d_in[2];
  const float* W1d   = (const float*)d_in[3];
  const float* att1s = (const float*)d_in[4];
  const float* att1d = (const float*)d_in[5];
  const float* b1    = (const float*)d_in[6];
  const float* W2s   = (const float*)d_in[7];
  const float* W2d   = (const float*)d_in[8];
  const float* att2s = (const float*)d_in[9];
  const float* att2d = (const float*)d_in[10];
  const float* b2    = (const float*)d_in[11];
  const float* fc1w  = (const float*)d_in[12];
  const float* fc1b  = (const float*)d_in[13];
  const float* gamma = (const float*)d_in[14];
  const float* beta  = (const float*)d_in[15];
  const float* fc2w  = (const float*)d_in[16];
  const float* fc2b  = (const float*)d_in[17];
  float* out = (float*)d_out;

  const int H1 = in_sizes[4];             // 128
  const int H2 = in_sizes[9];             // 128
  const int DI = in_sizes[2] / H1;        // 256
  const int N  = in_sizes[0] / DI;        // 100000
  const int E  = in_sizes[1] / 2;         // 1.6M
  const int Et = E + N;                   // + self loops
  const int FC = in_sizes[13];            // 64
  const int C  = in_sizes[17];            // 40

  // workspace carve-up (fp32)
  float* ws = (float*)d_ws;
  size_t off = 0;
  float* hsrc = ws + off; off += (size_t)N * H1;
  float* hdst = ws + off; off += (size_t)N * H1;
  float* hout = ws + off; off += (size_t)N * H1;
  float* esc  = ws + off; off += (size_t)Et;
  float* asrc = ws + off; off += N;
  float* adst = ws + off; off += N;
  float* mx   = ws + off; off += N;
  float* den  = ws + off; off += N;
  float* bnS  = ws + off; off += FC;      // sum
  float* bnQ  = ws + off; off += FC;      // sumsq
  float* bnA  = ws + off; off += FC;      // scale
  float* bnB  = ws + off; off += FC;      // shift
  float* fc1o = hsrc;                     // reuse: hsrc free after layer-2 scatter

  const int TB = 256, WPB = TB / 32;
  int rowTiles = divup(N, 16);
  dim3 blk(TB);

  // ---------------- GAT layer 1 ----------------
  gat_gemm_dual<<<divup(rowTiles * 2, WPB), blk, 0, stream>>>(x, W1s, W1d, hsrc, hdst, N, DI);
  gat_dots<<<divup(N, WPB), blk, 0, stream>>>(hsrc, hdst, att1s, att1d, asrc, adst, N, H1);
  init_stats_kernel<<<divup(N, TB), blk, 0, stream>>>(mx, den, N);
  fill_bias_kernel<<<divup((long long)N * H1, TB), blk, 0, stream>>>(hout, b1, (long long)N * H1, H1);
  edge_scores_kernel<<<divup(Et, TB), blk, 0, stream>>>(ei, asrc, adst, esc, mx, E, Et);
  edge_exp_kernel<<<divup(Et, TB), blk, 0, stream>>>(ei, esc, mx, den, E, Et);
  edge_scatter_kernel<<<divup(Et, WPB), blk, 0, stream>>>(ei, esc, den, hsrc, hout, E, Et);
  relu_kernel<<<divup((long long)N * H1, TB), blk, 0, stream>>>(hout, (long long)N * H1);

  // ---------------- GAT layer 2 ----------------
  gat_gemm_dual<<<divup(rowTiles * 2, WPB), blk, 0, stream>>>(hout, W2s, W2d, hsrc, hdst, N, H1);
  gat_dots<<<divup(N, WPB), blk, 0, stream>>>(hsrc, hdst, att2s, att2d, asrc, adst, N, H2);
  init_stats_kernel<<<divup(N, TB), blk, 0, stream>>>(mx, den, N);
  fill_bias_kernel<<<divup((long long)N * H2, TB), blk, 0, stream>>>(hout, b2, (long long)N * H2, H2);
  edge_scores_kernel<<<divup(Et, TB), blk, 0, stream>>>(ei, asrc, adst, esc, mx, E, Et);
  edge_exp_kernel<<<divup(Et, TB), blk, 0, stream>>>(ei, esc, mx, den, E, Et);
  edge_scatter_kernel<<<divup(Et, WPB), blk, 0, stream>>>(ei, esc, den, hsrc, hout, E, Et);
  relu_kernel<<<divup((long long)N * H2, TB), blk, 0, stream>>>(hout, (long long)N * H2);

  // ---------------- FC1 + BN + relu ----------------
  gemm_bias_wmma<4><<<divup(rowTiles, WPB), blk, 0, stream>>>(hout, fc1w, fc1b, fc1o, N, H2, FC);
  zero_kernel<<<1, 2 * FC, 0, stream>>>(bnS, 2 * FC);   // bnS & bnQ are contiguous
  bn_partial_kernel<<<512, FC, 0, stream>>>(fc1o, bnS, bnQ, N, FC);
  bn_finalize_kernel<<<1, FC, 0, stream>>>(bnS, bnQ, gamma, beta, bnA, bnB, N, FC);
  bn_apply_relu_kernel<<<divup((long long)N * FC, TB), blk, 0, stream>>>(fc1o, bnA, bnB, (long long)N * FC, FC);

  // ---------------- FC2 + log_softmax ----------------
  gemm_bias_wmma<3><<<divup(rowTiles, WPB), blk, 0, stream>>>(fc1o, fc2w, fc2b, out, N, FC, C);
  log_softmax_kernel<<<divup(N, WPB), blk, 0, stream>>>(out, N, C);
}